// SentencePairModel_48180943126860
// MI455X (gfx1250) — compile-verified
//
#include <hip/hip_runtime.h>
#include <limits.h>

typedef float v2f __attribute__((ext_vector_type(2)));
typedef float v8f __attribute__((ext_vector_type(8)));

#define NEG_INF (-1e30f)

// ---------------------------------------------------------------------------
// Kernel 1: GEMV logits = A @ W + b via V_WMMA_F32_16X16X4_F32.
// One wave handles 16 rows. A-tile: lane L supplies row (L&15), K-pair
// (2*(L>>4)) -> one global_load_b64 (non-temporal: A is stream-once, 205MB >
// 192MB L2) per lane per WMMA.  B is W replicated across all 16 columns
// (regular-temporal: 4KB reused by every wave), so every D column holds the
// same 16 dot products.
// ---------------------------------------------------------------------------
__global__ __launch_bounds__(256) void gemv_wmma_f32(
    const float* __restrict__ A, const float* __restrict__ W,
    const float* __restrict__ bias, float* __restrict__ out,
    int N, int Hdim) {
  int wave = (int)((blockIdx.x * blockDim.x + threadIdx.x) >> 5);
  int lane = (int)(threadIdx.x & 31);
  int row0 = wave * 16;
  if (row0 >= N) return;                   // uniform per wave

  int m = lane & 15;
  int khalf = (lane >> 4) * 2;             // 0 for lanes 0-15, 2 for 16-31
  int row = row0 + m;
  if (row >= N) row = N - 1;               // clamp padded rows (not stored)

  const float* arow = A + (size_t)row * (size_t)Hdim + khalf;
  const float* wp   = W + khalf;

  v8f acc0 = {}; v8f acc1 = {}; v8f acc2 = {}; v8f acc3 = {};
  for (int k = 0; k < Hdim; k += 16) {
    v2f a0 = __builtin_nontemporal_load((const v2f*)(arow + k));
    v2f b0 = *(const v2f*)(wp + k);
    v2f a1 = __builtin_nontemporal_load((const v2f*)(arow + k + 4));
    v2f b1 = *(const v2f*)(wp + k + 4);
    v2f a2 = __builtin_nontemporal_load((const v2f*)(arow + k + 8));
    v2f b2 = *(const v2f*)(wp + k + 8);
    v2f a3 = __builtin_nontemporal_load((const v2f*)(arow + k + 12));
    v2f b3 = *(const v2f*)(wp + k + 12);
    acc0 = __builtin_amdgcn_wmma_f32_16x16x4_f32(false, a0, false, b0,
                                                 (short)0, acc0, false, false);
    acc1 = __builtin_amdgcn_wmma_f32_16x16x4_f32(false, a1, false, b1,
                                                 (short)0, acc1, false, false);
    acc2 = __builtin_amdgcn_wmma_f32_16x16x4_f32(false, a2, false, b2,
                                                 (short)0, acc2, false, false);
    acc3 = __builtin_amdgcn_wmma_f32_16x16x4_f32(false, a3, false, b3,
                                                 (short)0, acc3, false, false);
  }
  v8f acc = (acc0 + acc1) + (acc2 + acc3);
  float bv = bias[0];

  // D layout: VGPR r, lanes 0-15 -> M=r ; lanes 16-31 -> M=8+r. N columns all
  // identical, so lane 0 writes rows row0..row0+7, lane 16 rows row0+8..+15.
  if ((lane & 15) == 0) {
    int mb = row0 + (lane >> 4) * 8;
#pragma unroll
    for (int r = 0; r < 8; ++r) {
      int rr = mb + r;
      if (rr < N) out[rr] = acc[r] + bv;
    }
  }
}

// ---------------------------------------------------------------------------
// Kernel 2: exclusive scan of lens -> offsets. Single wave, shfl-based.
// ---------------------------------------------------------------------------
__global__ void scan_offsets(const int* __restrict__ lens,
                             int* __restrict__ offsets, int S) {
  int lane = (int)threadIdx.x;
  int run = 0;
  for (int base = 0; base < S; base += 32) {
    int i = base + lane;
    int x = (i < S) ? lens[i] : 0;
    int v = x;
#pragma unroll
    for (int d = 1; d < 32; d <<= 1) {
      int t = __shfl_up(v, d, 32);
      if (lane >= d) v += t;
    }
    if (i < S) offsets[i] = run + (v - x);       // exclusive
    run += __shfl(v, 31, 32);
  }
}

// ---------------------------------------------------------------------------
// Kernel 3: per-segment double-softmax CE. One wave per segment, 2 elems/lane
// (MAX_LEN = 64). Writes per-segment loss + include flag to workspace.
// ---------------------------------------------------------------------------
__global__ __launch_bounds__(256) void seg_loss(
    const float* __restrict__ logits, const int* __restrict__ lens,
    const int* __restrict__ offsets, const int* __restrict__ tgt,
    float* __restrict__ loss_ws, float* __restrict__ inc_ws,
    int S, int EACH) {
  int wave = (int)((blockIdx.x * blockDim.x + threadIdx.x) >> 5);
  int lane = (int)(threadIdx.x & 31);
  int s = wave;
  if (s >= S) return;

  int len = lens[s];
  int off = offsets[s];
  int j0 = lane, j1 = lane + 32;
  bool v0 = j0 < len, v1 = j1 < len;
  float x0 = v0 ? logits[off + j0] : NEG_INF;
  float x1 = v1 ? logits[off + j1] : NEG_INF;
  int   t0 = v0 ? tgt[off + j0] : 0;
  int   t1 = v1 ? tgt[off + j1] : 0;

  // softmax over valid logits
  float mx = fmaxf(x0, x1);
#pragma unroll
  for (int d = 16; d > 0; d >>= 1) mx = fmaxf(mx, __shfl_xor(mx, d, 32));
  float e0 = v0 ? __expf(x0 - mx) : 0.0f;
  float e1 = v1 ? __expf(x1 - mx) : 0.0f;
  float sum = e0 + e1;
#pragma unroll
  for (int d = 16; d > 0; d >>= 1) sum += __shfl_xor(sum, d, 32);
  float inv = 1.0f / sum;
  float p0 = e0 * inv, p1 = e1 * inv;

  // second logsumexp over (valid ? p : -inf)
  float q0 = v0 ? p0 : NEG_INF;
  float q1 = v1 ? p1 : NEG_INF;
  float m2 = fmaxf(q0, q1);
#pragma unroll
  for (int d = 16; d > 0; d >>= 1) m2 = fmaxf(m2, __shfl_xor(m2, d, 32));
  float s2 = (v0 ? __expf(p0 - m2) : 0.0f) + (v1 ? __expf(p1 - m2) : 0.0f);
#pragma unroll
  for (int d = 16; d > 0; d >>= 1) s2 += __shfl_xor(s2, d, 32);
  float lse2 = m2 + __logf(s2);

  // argmax of (valid ? tgt : -1)  == first index with tgt==1, else 0
  int k0 = (v0 && t0 == 1) ? j0 : INT_MAX;
  int k1 = (v1 && t1 == 1) ? j1 : INT_MAX;
  int tk = min(k0, k1);
#pragma unroll
  for (int d = 16; d > 0; d >>= 1) tk = min(tk, __shfl_xor(tk, d, 32));
  int t_idx = (tk == INT_MAX) ? 0 : tk;
  float psel = (t_idx >= 32) ? p1 : p0;     // t_idx uniform across wave
  float p_t = __shfl(psel, t_idx & 31, 32);
  float loss_per = lse2 - p_t;

  // tgt_sum for skip_no_pos
  int ts = t0 + t1;
#pragma unroll
  for (int d = 16; d > 0; d >>= 1) ts += __shfl_xor(ts, d, 32);

  int next_len = (s + 1 < S) ? lens[s + 1] : 1;
  bool skip_next_zero = (s > 0) && (s + 1 < S) && (next_len == 0);
  bool skip_no_pos = (((s + 1) % EACH) == 0) && (ts <= 0);
  bool include = (len > 0) && !skip_next_zero && !skip_no_pos;

  if (lane == 0) {
    loss_ws[s] = include ? loss_per : 0.0f;
    inc_ws[s]  = include ? 1.0f : 0.0f;
  }
}

// ---------------------------------------------------------------------------
// Kernel 4: deterministic single-block reduction -> final scalar loss.
// ---------------------------------------------------------------------------
__global__ __launch_bounds__(256) void final_reduce(
    const float* __restrict__ loss_ws, const float* __restrict__ inc_ws,
    float* __restrict__ out_loss, int S) {
  __shared__ float sl[256];
  __shared__ float sc[256];
  int t = (int)threadIdx.x;
  float a = 0.0f, c = 0.0f;
  for (int i = t; i < S; i += 256) { a += loss_ws[i]; c += inc_ws[i]; }
  sl[t] = a; sc[t] = c;
  __syncthreads();
  for (int d = 128; d > 0; d >>= 1) {
    if (t < d) { sl[t] += sl[t + d]; sc[t] += sc[t + d]; }
    __syncthreads();
  }
  if (t == 0) {
    float cnt = sc[0];
    float denom = (cnt > 0.0f) ? cnt : 1e-4f;
    out_loss[0] = sl[0] / denom;
  }
}

// ---------------------------------------------------------------------------
extern "C" void kernel_launch(void* const* d_in, const int* in_sizes, int n_in,
                              void* d_out, int out_size, void* d_ws, size_t ws_size,
                              hipStream_t stream) {
  const float* A    = (const float*)d_in[0];   // cls_embedding [N,H]
  const float* W    = (const float*)d_in[1];   // [H]
  const float* b    = (const float*)d_in[2];   // [1]
  const int*   lens = (const int*)d_in[3];     // [S] (flattened [B,EACH])
  const int*   tgt  = (const int*)d_in[4];     // [N]

  int Hdim = in_sizes[1];
  int N    = in_sizes[0] / Hdim;
  int S    = in_sizes[3];
  const int EACH = 32;                         // length.shape[-1] in reference

  float* logits   = (float*)d_out;             // d_out = [logits(N), loss(1)]
  float* loss_out = logits + N;

  // workspace layout
  int*   offsets = (int*)d_ws;
  size_t obytes  = (((size_t)S * sizeof(int)) + 255) & ~(size_t)255;
  float* loss_ws = (float*)((char*)d_ws + obytes);
  float* inc_ws  = loss_ws + S;

  // 1) GEMV via fp32 WMMA: one wave per 16 rows, 8 waves per block
  int waves  = (N + 15) / 16;
  int blocks = (waves + 7) / 8;
  gemv_wmma_f32<<<blocks, 256, 0, stream>>>(A, W, b, logits, N, Hdim);

  // 2) offsets = exclusive cumsum(lens)
  scan_offsets<<<1, 32, 0, stream>>>(lens, offsets, S);

  // 3) per-segment loss (one wave per segment)
  seg_loss<<<(S + 7) / 8, 256, 0, stream>>>(logits, lens, offsets, tgt,
                                            loss_ws, inc_ws, S, EACH);

  // 4) final scalar loss
  final_reduce<<<1, 256, 0, stream>>>(loss_ws, inc_ws, loss_out, S);
}